// KalmanForecasterClosure3State_62646392980135
// MI455X (gfx1250) — compile-verified
//
#include <hip/hip_runtime.h>

// KalmanForecasterClosure3State for MI455X (gfx1250, wave32).
// One thread = one batch row; registers hold the 3-state + symmetric 3x3 P.
// CDNA5 data-movement paths: global_prefetch_b8 for history chunks,
// GLOBAL_LOAD_ASYNC_TO_LDS_B32 (ASYNCcnt) to stage the forecast tile into
// LDS while the 127-step history recurrence runs. WMMA is intentionally NOT
// used: per-lane independent 3x3 recurrences cannot map onto 16x16xK
// wave-level matrix ops (shared K would mix batch rows).

#define LH_  128
#define HF_  32
#define BLK  256

#if defined(__gfx1250__) && __has_builtin(__builtin_amdgcn_global_load_async_to_lds_b32) && __has_builtin(__builtin_amdgcn_s_wait_asynccnt)
#define HAVE_ASYNC_LDS 1
typedef __attribute__((address_space(1))) int* gptr_i32;
typedef __attribute__((address_space(3))) int* lptr_i32;
#else
#define HAVE_ASYNC_LDS 0
#endif

__device__ __forceinline__ float softplus_f(float x) {
    return (x > 15.0f) ? x : log1pf(expf(x));
}

__global__ __launch_bounds__(BLK)
void kf3_forecast_kernel(const float* __restrict__ v_hist,
                         const float* __restrict__ dt_hist,
                         const float* __restrict__ x_obs,
                         const float* __restrict__ v_fut,
                         const float* __restrict__ dt_fut,
                         const float* __restrict__ theta,
                         float* __restrict__ out, int Bn)
{
    __shared__ float s_vf[HF_ * BLK];   // [i][tid], conflict-free reads
    __shared__ float s_df[HF_ * BLK];

    const int tid = threadIdx.x;
    const int b   = blockIdx.x * BLK + tid;

    // ---- Stage forecast tile into LDS (overlaps with history compute) ----
    {
        const float* gv = v_fut  + (size_t)b * HF_;
        const float* gd = dt_fut + (size_t)b * HF_;
#if HAVE_ASYNC_LDS
        #pragma unroll
        for (int i = 0; i < HF_; ++i) {
            __builtin_amdgcn_global_load_async_to_lds_b32(
                (gptr_i32)(gv + i),
                (lptr_i32)&s_vf[i * BLK + tid], 0, 0);
            __builtin_amdgcn_global_load_async_to_lds_b32(
                (gptr_i32)(gd + i),
                (lptr_i32)&s_df[i * BLK + tid], 0, 0);
        }
#else
        #pragma unroll 4
        for (int i = 0; i < HF_; ++i) {
            s_vf[i * BLK + tid] = gv[i];
            s_df[i * BLK + tid] = gd[i];
        }
#endif
    }

    // ---- Derived parameters (uniform; scalar-cached loads) ----
    const float alpha = 1.0f / (1.0f + expf(-theta[0]));
    const float c_    = softplus_f(theta[1]);
    const float kappa = softplus_f(theta[2]);
    const float vc    = softplus_f(theta[3]);           // VC_MIN == 0
    const float vcsq  = vc * vc;
    const float qxs   = expf(theta[7]) * expf(theta[4]); // qs*qx
    const float qus   = expf(theta[7]) * expf(theta[5]); // qs*qu
    const float Rn    = expf(theta[6]);
    const float p0xx  = expf(theta[8]);
    const float p0uu  = expf(theta[9]);
    const float a1    = softplus_f(theta[10]);
    const float d1    = softplus_f(theta[11]);
    const float d2    = softplus_f(theta[12]);
    const float d3    = softplus_f(theta[13]);
    const float b1    = theta[14];
    const float b2    = theta[15];
    const float beta  = theta[16];
    const float rho_m = tanhf(theta[17]);
    const float qm    = expf(theta[18]);
    const float p0mm  = expf(theta[19]);

    // ---- Per-row state ----
    const float4* vr  = (const float4*)(v_hist  + (size_t)b * LH_);
    const float4* dtr = (const float4*)(dt_hist + (size_t)b * LH_);
    const float4* yr  = (const float4*)(x_obs   + (size_t)b * LH_);

    float x, u = 0.0f, m = 0.0f;
    float p00 = p0xx, p01 = 0.0f, p02 = 0.0f, p11 = p0uu, p12 = 0.0f, p22 = p0mm;

    // KF predict: state + symmetric covariance (F P F^T + Q)
    auto predict = [&](float v, float dv, float dtraw) {
        float dt  = fmaxf(dtraw, 1e-6f);
        float rho = __expf(-alpha * dt);
        float forcing = fmaf(v, v, -vcsq);
        forcing = forcing > 0.0f ? forcing : 0.0f;
        float damp = fmaf(d1, u, fmaf(d2, fabsf(v), fmaf(d3, fabsf(u), a1)));
        float cl   = fmaf(b1, v, fmaf(b2, dv, -u * damp));
        float w    = fmaf(c_, forcing, cl) - kappa * x;
        float x_p  = fmaf(u, dt, x);
        float u_p  = fmaf(rho, u, fmaf(dt, w, beta * m));
        float m_p  = rho_m * m;

        float f10  = -kappa * dt;
        float FP00 = fmaf(dt, p01, p00);
        float FP01 = fmaf(dt, p11, p01);
        float FP02 = fmaf(dt, p12, p02);
        float FP10 = fmaf(f10, p00, fmaf(rho, p01, beta * p02));
        float FP11 = fmaf(f10, p01, fmaf(rho, p11, beta * p12));
        float FP12 = fmaf(f10, p02, fmaf(rho, p12, beta * p22));

        float np00 = fmaf(dt, FP01, FP00) + qxs * dt;
        float np01 = fmaf(f10, FP00, fmaf(rho, FP01, beta * FP02));
        float np11 = fmaf(f10, FP10, fmaf(rho, FP11, beta * FP12)) + qus * dt;
        float np02 = rho_m * FP02;
        float np12 = rho_m * FP12;
        float np22 = fmaf(rho_m * rho_m, p22, qm);

        x = x_p; u = u_p; m = m_p;
        p00 = np00; p01 = np01; p02 = np02; p11 = np11; p12 = np12; p22 = np22;
    };

    // KF measurement update (Joseph form, symmetric)
    auto update = [&](float y) {
        float innov = y - x;
        float inv   = 1.0f / (p00 + Rn);
        float k0 = p00 * inv, k1 = p01 * inv, k2 = p02 * inv;
        x = fmaf(k0, innov, x);
        u = fmaf(k1, innov, u);
        m = fmaf(k2, innov, m);
        float omk  = 1.0f - k0;
        float MP00 = omk * p00, MP01 = omk * p01, MP02 = omk * p02;
        float MP10 = fmaf(-k1, p00, p01), MP11 = fmaf(-k1, p01, p11), MP12 = fmaf(-k1, p02, p12);
        float MP20 = fmaf(-k2, p00, p02), MP22 = fmaf(-k2, p02, p22);
        float Rk0 = Rn * k0, Rk1 = Rn * k1, Rk2 = Rn * k2;
        float np00 = fmaf(Rk0, k0, omk * MP00);
        float np01 = fmaf(Rk0, k1, fmaf(-k1, MP00, MP01));
        float np02 = fmaf(Rk0, k2, fmaf(-k2, MP00, MP02));
        float np11 = fmaf(Rk1, k1, fmaf(-k1, MP10, MP11));
        float np12 = fmaf(Rk1, k2, fmaf(-k2, MP10, MP12));
        float np22 = fmaf(Rk2, k2, fmaf(-k2, MP20, MP22));
        p00 = np00; p01 = np01; p02 = np02; p11 = np11; p12 = np12; p22 = np22;
    };

    // ---- History scan: 127 steps; step i uses v[i], dt[i+1], y[i+1] ----
    float4 dtc = dtr[0];
    float4 yc  = yr[0];
    x = yc.x;                                  // s0.x = x_obs[b,0]
    float vprev = ((const float*)vr)[0];       // makes dv(i=0) == 0

    #pragma unroll 1
    for (int c = 0; c < 31; ++c) {
        float4 vc4 = vr[c];
        float4 dtn = dtr[c + 1];
        float4 yn  = yr[c + 1];
        __builtin_prefetch((const void*)(vr  + c + 2), 0, 1);
        __builtin_prefetch((const void*)(dtr + c + 2), 0, 1);
        __builtin_prefetch((const void*)(yr  + c + 2), 0, 1);

        predict(vc4.x, vc4.x - vprev, dtc.y); update(yc.y);
        predict(vc4.y, vc4.y - vc4.x, dtc.z); update(yc.z);
        predict(vc4.z, vc4.z - vc4.y, dtc.w); update(yc.w);
        predict(vc4.w, vc4.w - vc4.z, dtn.x); update(yn.x);

        vprev = vc4.w; dtc = dtn; yc = yn;
    }
    // tail: steps 124..126 (dtc/yc now hold columns 124..127)
    float4 vt = vr[31];
    predict(vt.x, vt.x - vprev, dtc.y); update(yc.y);
    predict(vt.y, vt.y - vt.x,  dtc.z); update(yc.z);
    predict(vt.z, vt.z - vt.y,  dtc.w); update(yc.w);
    float vlast = vt.w;                        // v_hist[b, 127]

    // ---- Drain async staging (per-wave; each thread reads its own data) ----
#if HAVE_ASYNC_LDS
    __builtin_amdgcn_s_wait_asynccnt(0);
#endif

    // ---- Forecast: 32 predict-only steps from LDS tile ----
    float* ox = out + (size_t)b * HF_;
    float* ov = out + (size_t)Bn * HF_ + (size_t)b * HF_;
    float* ou = out + 2 * (size_t)Bn * HF_ + (size_t)b * HF_;

    float vp = vlast;
    #pragma unroll 1
    for (int c = 0; c < HF_ / 4; ++c) {
        float4 xo, po, uo;
        int i0 = 4 * c;
        float v0 = s_vf[(i0 + 0) * BLK + tid];
        predict(v0, v0 - vp, s_df[(i0 + 0) * BLK + tid]);
        xo.x = x; po.x = p00; uo.x = u;
        float v1 = s_vf[(i0 + 1) * BLK + tid];
        predict(v1, v1 - v0, s_df[(i0 + 1) * BLK + tid]);
        xo.y = x; po.y = p00; uo.y = u;
        float v2 = s_vf[(i0 + 2) * BLK + tid];
        predict(v2, v2 - v1, s_df[(i0 + 2) * BLK + tid]);
        xo.z = x; po.z = p00; uo.z = u;
        float v3 = s_vf[(i0 + 3) * BLK + tid];
        predict(v3, v3 - v2, s_df[(i0 + 3) * BLK + tid]);
        xo.w = x; po.w = p00; uo.w = u;
        vp = v3;
        ((float4*)ox)[c] = xo;
        ((float4*)ov)[c] = po;
        ((float4*)ou)[c] = uo;
    }
}

extern "C" void kernel_launch(void* const* d_in, const int* in_sizes, int n_in,
                              void* d_out, int out_size, void* d_ws, size_t ws_size,
                              hipStream_t stream) {
    const float* v_hist  = (const float*)d_in[0];
    const float* dt_hist = (const float*)d_in[1];
    const float* x_obs   = (const float*)d_in[2];
    const float* v_fut   = (const float*)d_in[3];
    const float* dt_fut  = (const float*)d_in[4];
    const float* theta   = (const float*)d_in[5];
    const int Bn = in_sizes[0] / LH_;          // 65536
    dim3 grid(Bn / BLK), block(BLK);
    kf3_forecast_kernel<<<grid, block, 0, stream>>>(v_hist, dt_hist, x_obs,
                                                    v_fut, dt_fut, theta,
                                                    (float*)d_out, Bn);
}